// CrossAttention_67929202753778
// MI455X (gfx1250) — compile-verified
//
#include <hip/hip_runtime.h>
#include <math.h>

// ---------------------------------------------------------------------------
// CrossAttention for MI455X (gfx1250, wave32, WMMA, 320KB-LDS WGP).
// Shapes: B=2, H=8, N=3072 (48x64), C=256, d=32. Output [B,H,N,N+d] f32.
// Matmuls use V_WMMA_F32_16X16X4_F32 (exact f32, matches f32 reference).
// Scores+softmax+AV fused per 16-query-row strip held in LDS: attn matrix is
// written to HBM exactly once and never re-read.
// ---------------------------------------------------------------------------

typedef __attribute__((ext_vector_type(2))) float v2f;
typedef __attribute__((ext_vector_type(8))) float v8f;

#define BB 2
#define HH 8
#define NN 3072
#define CC 256
#define DD 32
#define OUTW (NN + DD)        // 3104
#define VPW 48                // v2p padded width (d+6=38 -> 48)
#define KPF 304               // H*(d+6)
#define PITCH 3076            // LDS strip row pitch (pitch%64==4 -> no bank conflicts)

__device__ __forceinline__ v8f wmma_f32_k4(v2f a, v2f b, v8f c) {
  // (neg_a, A, neg_b, B, c_mod, C, reuse_a, reuse_b)
  return __builtin_amdgcn_wmma_f32_16x16x4_f32(false, a, false, b, (short)0, c,
                                               false, false);
}

// ---------------------------------------------------------------------------
// Generic row-major NN GEMM: C[M,Nn] = A[M,K] * B[K,Nn].  One 16x16 tile/wave.
// ---------------------------------------------------------------------------
__global__ void gemm_nn(const float* __restrict__ A, const float* __restrict__ B,
                        float* __restrict__ C, int M, int Nn, int K) {
  const int wave   = blockIdx.x * (blockDim.x >> 5) + (threadIdx.x >> 5);
  const int tilesN = Nn >> 4;
  const int total  = (M >> 4) * tilesN;
  if (wave >= total) return;                      // wave-uniform exit
  const int tm = wave / tilesN, tn = wave % tilesN;
  const int lane = threadIdx.x & 31;
  const int half = lane >> 4;
  const int l    = lane & 15;

  const float* Arow = A + (long)(tm * 16 + l) * K;
  const float* Bcol = B + (tn * 16 + l);
  v8f acc = {0.f, 0.f, 0.f, 0.f, 0.f, 0.f, 0.f, 0.f};

  for (int kk = 0; kk < K; kk += 4) {
    const int k0 = kk + 2 * half;
    v2f a, b;
    a.x = Arow[k0];
    a.y = Arow[k0 + 1];
    b.x = Bcol[(long)k0 * Nn];
    b.y = Bcol[(long)(k0 + 1) * Nn];
    acc = wmma_f32_k4(a, b, acc);
  }
  const int n = tn * 16 + l;
#pragma unroll
  for (int i = 0; i < 8; ++i) {
    const int m = tm * 16 + i + 8 * half;
    C[(long)m * Nn + n] = acc[i];
  }
}

// ---------------------------------------------------------------------------
// Build v2p [16 (=b*8+h), N, 48]: cols 0..31 = V head slice, 32..37 = pos, pad 0.
// ---------------------------------------------------------------------------
__global__ void build_vp(const float* __restrict__ KV, float* __restrict__ Vp) {
  const long idx = (long)blockIdx.x * blockDim.x + threadIdx.x;
  const long total = (long)BB * HH * NN * VPW;
  if (idx >= total) return;
  const int j  = (int)(idx % VPW);
  const long r = idx / VPW;
  const int n  = (int)(r % NN);
  const int bh = (int)(r / NN);
  const int b  = bh >> 3, h = bh & 7;

  float val;
  if (j < DD) {
    val = KV[((long)b * NN + n) * (2 * CC) + CC + h * DD + j];
  } else if (j < DD + 6) {
    const int jj = j - DD;
    const float p3 = -1.0f + 2.0f * (float)(n % 48) / 47.0f;  // ys tiled
    const float p4 = -1.0f + 2.0f * (float)(n / 48) / 63.0f;  // xs repeated
    float pv[6] = {p3 * p3, p4 * p4, p3 * p4, p3, p4, 1.0f};
    val = pv[jj];
  } else {
    val = 0.0f;
  }
  Vp[idx] = val;
}

// ---------------------------------------------------------------------------
// Fused scores + softmax + AV for one 16-query-row strip.
//   grid = (192, 16);  block = 256 threads (8 waves);  ~216KB static LDS.
// Phase 1: strip[16][3072] = Q_tile * K^T * scale      (WMMA, K=32)
// Phase 2: row softmax in LDS, normalized attn written once to d_out
// Phase 3: T[16][48] = strip * v2p  (WMMA, K split 384/wave, LDS A operands)
// ---------------------------------------------------------------------------
__global__ __launch_bounds__(256) void fused_attn(
    const float* __restrict__ Q, const float* __restrict__ KV,
    const float* __restrict__ Vp, float* __restrict__ out,
    float* __restrict__ T) {
  __shared__ float strip[16 * PITCH];     // 196,864 B
  __shared__ float part[8 * 16 * VPW];    //  24,576 B

  const int bh = blockIdx.y;
  const int b = bh >> 3, h = bh & 7;
  const int tm = blockIdx.x;              // 16-row strip index (0..191)
  const int w = threadIdx.x >> 5;         // wave id (0..7)
  const int lane = threadIdx.x & 31;
  const int half = lane >> 4;
  const int l = lane & 15;

  // ---- Phase 1: logits into LDS strip --------------------------------------
  {
    const float* Qb = Q + (long)b * NN * CC + h * DD;
    const float* Kb = KV + (long)b * NN * (2 * CC) + h * DD;
    const float* Arow = Qb + (long)(tm * 16 + l) * CC;
    const float scale = 0.17677669529663687f;  // 32^-0.5

    for (int tn = w; tn < (NN >> 4); tn += 8) {
      const float* Brow = Kb + (long)(tn * 16 + l) * (2 * CC);
      v8f acc = {0.f, 0.f, 0.f, 0.f, 0.f, 0.f, 0.f, 0.f};
#pragma unroll
      for (int kk = 0; kk < DD; kk += 4) {
        const int k0 = kk + 2 * half;
        v2f a, bv;
        a.x = Arow[k0];  a.y = Arow[k0 + 1];
        bv.x = Brow[k0]; bv.y = Brow[k0 + 1];
        acc = wmma_f32_k4(a, bv, acc);
      }
      const int n = tn * 16 + l;
#pragma unroll
      for (int i = 0; i < 8; ++i)
        strip[(i + 8 * half) * PITCH + n] = acc[i] * scale;
    }
  }
  __syncthreads();

  // ---- Phase 2: softmax per row (wave handles 2 rows), write attn once -----
  for (int rr = 0; rr < 2; ++rr) {
    const int r = w * 2 + rr;
    float* rowp = &strip[r * PITCH];
    float m = -3.402823466e38f;
    for (int i = lane; i < NN; i += 32) m = fmaxf(m, rowp[i]);
#pragma unroll
    for (int off = 16; off > 0; off >>= 1) m = fmaxf(m, __shfl_xor(m, off, 32));
    float s = 0.0f;
    for (int i = lane; i < NN; i += 32) {
      const float e = expf(rowp[i] - m);
      rowp[i] = e;
      s += e;
    }
#pragma unroll
    for (int off = 16; off > 0; off >>= 1) s += __shfl_xor(s, off, 32);
    const float inv = 1.0f / s;
    float* gout = out + ((long)bh * NN + tm * 16 + r) * OUTW;
    for (int i = lane; i < NN; i += 32) {
      const float v = rowp[i] * inv;
      rowp[i] = v;       // keep normalized attn for phase 3
      gout[i] = v;       // single HBM write of the attn matrix
    }
  }
  __syncthreads();

  // ---- Phase 3: T strip = attn_strip * v2p (K split across 8 waves) --------
  {
    const float* Bb = Vp + (long)bh * NN * VPW;
    v8f acc0 = {0.f, 0.f, 0.f, 0.f, 0.f, 0.f, 0.f, 0.f};
    v8f acc1 = acc0, acc2 = acc0;
    const int kbeg = w * (NN / 8), kend = kbeg + (NN / 8);
    for (int kk = kbeg; kk < kend; kk += 4) {
      const int k0 = kk + 2 * half;
      v2f a, b0, b1, b2;
      a.x = strip[l * PITCH + k0];
      a.y = strip[l * PITCH + k0 + 1];
      const float* B0 = Bb + (long)k0 * VPW;
      const float* B1 = Bb + (long)(k0 + 1) * VPW;
      b0.x = B0[l];      b0.y = B1[l];
      b1.x = B0[16 + l]; b1.y = B1[16 + l];
      b2.x = B0[32 + l]; b2.y = B1[32 + l];
      acc0 = wmma_f32_k4(a, b0, acc0);
      acc1 = wmma_f32_k4(a, b1, acc1);
      acc2 = wmma_f32_k4(a, b2, acc2);
    }
#pragma unroll
    for (int i = 0; i < 8; ++i) {
      const int m = i + 8 * half;
      part[(w * 16 + m) * VPW + l]      = acc0[i];
      part[(w * 16 + m) * VPW + 16 + l] = acc1[i];
      part[(w * 16 + m) * VPW + 32 + l] = acc2[i];
    }
  }
  __syncthreads();

  // Reduce 8 wave-partials and store T strip
  for (int e = threadIdx.x; e < 16 * VPW; e += 256) {
    float s = 0.0f;
#pragma unroll
    for (int ww = 0; ww < 8; ++ww) s += part[ww * 16 * VPW + e];
    T[((long)bh * NN + tm * 16 + (e / VPW)) * VPW + (e % VPW)] = s;
  }
}

// ---------------------------------------------------------------------------
// y = T_flat[B*N, 304] * Wpf[304, 256] + bpf, scattered into
// out[b, h', n, 3072 + dd] with h'=col>>5, dd=col&31.
// ---------------------------------------------------------------------------
__global__ void final_proj(const float* __restrict__ T,
                           const float* __restrict__ Wpf,
                           const float* __restrict__ bpf,
                           float* __restrict__ out) {
  const int wave = blockIdx.x * (blockDim.x >> 5) + (threadIdx.x >> 5);
  const int tilesN = CC >> 4;                                // 16
  const int total = ((BB * NN) >> 4) * tilesN;               // 6144
  if (wave >= total) return;
  const int tm = wave / tilesN, tn = wave % tilesN;
  const int lane = threadIdx.x & 31;
  const int half = lane >> 4;
  const int l    = lane & 15;

  const int row = tm * 16 + l;                               // (b, n)
  const int b = row / NN, n = row % NN;
  const float* Wcol = Wpf + (tn * 16 + l);
  v8f acc = {0.f, 0.f, 0.f, 0.f, 0.f, 0.f, 0.f, 0.f};

  for (int kk = 0; kk < KPF; kk += 4) {
    const int k0 = kk + 2 * half;
    const int k1 = k0 + 1;
    const int h0 = k0 / 38, j0 = k0 % 38;
    const int h1 = k1 / 38, j1 = k1 % 38;
    v2f a, bv;
    a.x = T[((long)(b * HH + h0) * NN + n) * VPW + j0];
    a.y = T[((long)(b * HH + h1) * NN + n) * VPW + j1];
    bv.x = Wcol[(long)k0 * CC];
    bv.y = Wcol[(long)k1 * CC];
    acc = wmma_f32_k4(a, bv, acc);
  }
  const int c = tn * 16 + l;                                 // output column
  const int hp = c >> 5, dd = c & 31;
  const float bias = bpf[c];
#pragma unroll
  for (int i = 0; i < 8; ++i) {
    const int m = tm * 16 + i + 8 * half;                    // global row
    const int bb = m / NN, nn2 = m % NN;
    out[((long)(bb * HH + hp) * NN + nn2) * OUTW + NN + dd] = acc[i] + bias;
  }
}

// ---------------------------------------------------------------------------
extern "C" void kernel_launch(void* const* d_in, const int* in_sizes, int n_in,
                              void* d_out, int out_size, void* d_ws,
                              size_t ws_size, hipStream_t stream) {
  const float* x1  = (const float*)d_in[0];   // [2,3072,256]
  const float* x2  = (const float*)d_in[1];   // [2,3072,256]
  const float* Wq  = (const float*)d_in[2];   // [256,256]
  const float* Wkv = (const float*)d_in[3];   // [256,512]
  const float* Wpf = (const float*)d_in[4];   // [304,256]
  const float* bpf = (const float*)d_in[5];   // [256]
  float* out = (float*)d_out;                 // [2,8,3072,3104]

  // Workspace layout (floats)
  float* Qbuf  = (float*)d_ws;                           // 6144*256
  float* KVbuf = Qbuf + (long)BB * NN * CC;              // 6144*512
  float* Vp    = KVbuf + (long)BB * NN * 2 * CC;         // 16*3072*48
  float* Tbuf  = Vp + (long)BB * HH * NN * VPW;          // 16*3072*48

  const int M = BB * NN;  // 6144

  // 1) Q = x1 @ Wq   (6144 x 256 x 256)
  gemm_nn<<<dim3((M / 16) * (CC / 16) / 4), dim3(128), 0, stream>>>(
      x1, Wq, Qbuf, M, CC, CC);

  // 2) KV = x2 @ Wkv (6144 x 512 x 256)
  gemm_nn<<<dim3((M / 16) * (2 * CC / 16) / 4), dim3(128), 0, stream>>>(
      x2, Wkv, KVbuf, M, 2 * CC, CC);

  // 3) Build v2p (with positional encodings, padded to 48)
  {
    const long total = (long)BB * HH * NN * VPW;
    build_vp<<<dim3((unsigned)((total + 255) / 256)), dim3(256), 0, stream>>>(
        KVbuf, Vp);
  }

  // 4) Fused scores + softmax + AV: one block per 16-row strip
  fused_attn<<<dim3(NN / 16, BB * HH), dim3(256), 0, stream>>>(Qbuf, KVbuf, Vp,
                                                               out, Tbuf);

  // 5) y = T_flat @ Wpf + bpf, scattered into d_out y region
  final_proj<<<dim3((M / 16) * (CC / 16) / 4), dim3(128), 0, stream>>>(
      Tbuf, Wpf, bpf, out);
}